// CFNOTF_80066780332643
// MI455X (gfx1250) — compile-verified
//
#include <hip/hip_runtime.h>

typedef __attribute__((ext_vector_type(2))) float v2f;
typedef __attribute__((ext_vector_type(8))) float v8f;

#define TWO_PI 6.2831853071795864769f
#define EPSV 1e-5f

// ---------------- Stage A1: A[n,d], B[n,d] spectral sums (n<256, d<16) -------
__global__ void cfno_stageA1(const float* __restrict__ Wr, const float* __restrict__ Wi,
                             float* __restrict__ Aarr, float* __restrict__ Barr) {
    int tid = blockIdx.x * blockDim.x + threadIdx.x;   // 4096 threads
    int n = tid >> 4, d = tid & 15;
    float sa = 0.f, sb = 0.f;
    for (int k = 0; k < 256; ++k) {
        int ph = (k * n) & 255;                         // exact angle mod 2*pi
        float th = (float)ph * (TWO_PI / 256.0f);
        float sn, cs;
        __sincosf(th, &sn, &cs);
        float wr = Wr[k * 16 + d], wi = Wi[k * 16 + d];
        sa += cs * wr + sn * wi;                        // coeff of p_n in yr_d
        sb += cs * wi - sn * wr;                        // coeff of p_n in yi_d
    }
    Aarr[tid] = sa;
    Barr[tid] = sb;
}

// ---- Stage A2: fold IFFT16 -> M[n,m]; emit in WMMA B-fragment layout + c ----
// Fragment layout for V_WMMA_F32_16X16X4_F32 B (4x16):
//   iteration it (K-chunk of 4), lane = 16*(K pair hi) + n, comp j in {0,1}
//   holds B[4*it + 2*hi + j][n]
__global__ void cfno_stageA2(const float* __restrict__ Aarr, const float* __restrict__ Barr,
                             const float* __restrict__ br, const float* __restrict__ bi,
                             float* __restrict__ Bbuf, float* __restrict__ cvec) {
    int tid = blockIdx.x * blockDim.x + threadIdx.x;   // 4096 threads
    int n = tid >> 4, m = tid & 15;
    float s = 0.f;
    for (int d = 0; d < 16; ++d) {
        int ph = (d * m) & 15;
        float phi = (float)ph * (TWO_PI / 16.0f);
        float sn, cs;
        __sincosf(phi, &sn, &cs);
        s += Aarr[n * 16 + d] * cs - Barr[n * 16 + d] * sn;
    }
    s *= (1.0f / 16.0f);
    int it = n >> 2, within = n & 3;
    int lane = ((within >> 1) << 4) + m;               // 16*hi + n(channel)
    Bbuf[(it * 32 + lane) * 2 + (within & 1)] = s;
    if (n == 0) {
        float c = 0.f;
        for (int d = 0; d < 16; ++d) {
            int ph = (d * m) & 15;
            float phi = (float)ph * (TWO_PI / 16.0f);
            float sn, cs;
            __sincosf(phi, &sn, &cs);
            c += (br[d] - bi[d]) * cs - (br[d] + bi[d]) * sn;
        }
        cvec[m] = c * (1.0f / 16.0f);
    }
}

// ------------- Stage B: [131072 x 256] @ [256 x 16] via fp32 WMMA ------------
// One wave -> 16 patches (one patch-row group) x 16 channels.
__global__ void cfno_stageB(const float* __restrict__ x, const float* __restrict__ Bbuf,
                            const float* __restrict__ cvec, float* __restrict__ y) {
    int lane = threadIdx.x & 31;
    int wave = threadIdx.x >> 5;
    int gw   = blockIdx.x * (blockDim.x >> 5) + wave;  // 0..8191
    int b    = gw >> 12;
    int rem  = gw & 4095;
    int ph   = rem >> 4;                               // patch row 0..255
    int pw0  = (rem & 15) << 4;                        // patch col group base
    int ml   = lane & 15;                              // A: patch idx / B: channel
    int khi  = lane >> 4;                              // selects K pair (0: k,k+1  1: k+2,k+3)

    const float* base = x + (((size_t)b * 4096 + (size_t)ph * 16) * 4096)
                          + (size_t)(pw0 + ml) * 16;
    const v2f* Bf = (const v2f*)Bbuf;

    v8f acc = (v8f)(0.0f);
#pragma unroll 4
    for (int it = 0; it < 64; ++it) {
        int k   = (it << 2) + (khi << 1);              // element index in patch
        int row = k >> 4, col = k & 15;                // col even -> 8B aligned
        v2f a = *(const v2f*)(base + (size_t)row * 4096 + col);
        v2f bf = Bf[it * 32 + lane];                   // coalesced, L2-resident
        acc = __builtin_amdgcn_wmma_f32_16x16x4_f32(
                  /*neg_a=*/false, a, /*neg_b=*/false, bf,
                  /*c_mod=*/(short)0, acc, /*reuse_a=*/false, /*reuse_b=*/false);
    }

    float cb = cvec[ml];
    size_t ybase = (((size_t)b * 256 + ph) * 256 + pw0) * 16;
#pragma unroll
    for (int r = 0; r < 8; ++r) {
        int patch = r + (khi << 3);                    // C/D layout: vgpr r, lane-half
        y[ybase + (size_t)patch * 16 + ml] = acc[r] + cb;
    }
}

// ------- Stage C: 3x3 depthwise conv + BN, only at odd (r,c) = resize --------
__global__ void cfno_stageC(const float* __restrict__ y, const float* __restrict__ dwk,
                            const float* __restrict__ dwb, const float* __restrict__ gamma,
                            const float* __restrict__ beta, const float* __restrict__ mmean,
                            const float* __restrict__ mvar, float* __restrict__ out) {
    int idx = blockIdx.x * blockDim.x + threadIdx.x;   // 2*128*128*16 = 524288
    int d  = idx & 15;
    int ow = (idx >> 4) & 127;
    int oh = (idx >> 11) & 127;
    int b  = idx >> 18;
    int r = 2 * oh + 1, c = 2 * ow + 1;                // nearest half-pixel centers
    float sum = 0.f;
#pragma unroll
    for (int dy = -1; dy <= 1; ++dy) {
        int rr = r + dy;
        if (rr < 0 || rr > 255) continue;              // SAME zero pad
#pragma unroll
        for (int dx = -1; dx <= 1; ++dx) {
            int cc = c + dx;
            if (cc < 0 || cc > 255) continue;
            float v = y[(((size_t)b * 256 + rr) * 256 + cc) * 16 + d];
            float w = dwk[((size_t)(dy + 1) * 3 + (dx + 1)) * 16 + d];
            sum += v * w;
        }
    }
    float v = sum + dwb[d];
    float sc = gamma[d] * rsqrtf(mvar[d] + EPSV);
    out[idx] = sc * (v - mmean[d]) + beta[d];
}

extern "C" void kernel_launch(void* const* d_in, const int* in_sizes, int n_in,
                              void* d_out, int out_size, void* d_ws, size_t ws_size,
                              hipStream_t stream) {
    (void)in_sizes; (void)n_in; (void)out_size; (void)ws_size;
    const float* x     = (const float*)d_in[0];
    const float* Wr    = (const float*)d_in[1];
    const float* br    = (const float*)d_in[2];
    const float* Wi    = (const float*)d_in[3];
    const float* bi    = (const float*)d_in[4];
    const float* dwk   = (const float*)d_in[5];
    const float* dwb   = (const float*)d_in[6];
    const float* gamma = (const float*)d_in[7];
    const float* beta  = (const float*)d_in[8];
    const float* mmean = (const float*)d_in[9];
    const float* mvar  = (const float*)d_in[10];
    float* out = (float*)d_out;

    char* ws = (char*)d_ws;
    float* Aarr = (float*)ws;                   // 4096 f
    float* Barr = Aarr + 4096;                  // 4096 f
    float* Bbuf = Barr + 4096;                  // 4096 f (fragment-layout M)
    float* cvec = Bbuf + 4096;                  // 16 f
    float* y    = (float*)(ws + 65536);         // 2*256*256*16 f = 8 MiB

    cfno_stageA1<<<16, 256, 0, stream>>>(Wr, Wi, Aarr, Barr);
    cfno_stageA2<<<16, 256, 0, stream>>>(Aarr, Barr, br, bi, Bbuf, cvec);
    cfno_stageB<<<1024, 256, 0, stream>>>(x, Bbuf, cvec, y);
    cfno_stageC<<<2048, 256, 0, stream>>>(y, dwk, dwb, gamma, beta, mmean, mvar, out);
}